// Conv2dIm2Col_34291018891444
// MI455X (gfx1250) — compile-verified
//
#include <hip/hip_runtime.h>

typedef float v2f __attribute__((ext_vector_type(2)));
typedef float v8f __attribute__((ext_vector_type(8)));
typedef unsigned int v4u __attribute__((ext_vector_type(4)));
typedef int v4i __attribute__((ext_vector_type(4)));
typedef int v8i __attribute__((ext_vector_type(8)));

#define BATCH   16
#define CIN     64
#define HW      112
#define COUT    64
#define PIX     (HW * HW)     // 12544
#define PIX_PER_WG 256
#define CHUNKS  8             // 8 chunks x (2 n-waves x 16 pixels) = 256 pixels/WG

// 8 waves: 4 M-waves (16 couts each) x 2 N-waves (16 pixels each)
__launch_bounds__(256)
__global__ void conv3x3_im2col_wmma_f32(const float* __restrict__ x,
                                        const float* __restrict__ w,
                                        const float* __restrict__ bias,
                                        float* __restrict__ out) {
  // Weights in LDS as [tap t][cin c][cout m] : 576*64 f32 = 147456 B.
  __shared__ float lw[9 * CIN * COUT];
  __shared__ float lb[COUT];

  const int tid = threadIdx.x;
  const int wid = tid >> 5;

#if __has_builtin(__builtin_amdgcn_tensor_load_to_lds) && __has_builtin(__builtin_amdgcn_s_wait_tensorcnt)
  // ---- TDM: one 3-D tensor DMA does the whole [c][t][m] -> [t][c][m]
  //      transposed weight staging into LDS (source w is [k= c*9+t][m]).
  //      Dest iterates X=m (64, contiguous), Y=c (64, src stride 576),
  //      Z=t (9, src stride 64); LDS filled contiguously => [t][c][m].
  if (wid == 0) {   // uniform branch: only wave 0 issues the DMA
    const unsigned long long ga = (unsigned long long)(uintptr_t)w;

    v4u g0;
    g0.x = 1u;                                   // count=1, user desc, no gather
    g0.y = (unsigned)(uintptr_t)lw;              // lds_addr (bytes)
    g0.z = (unsigned)(ga & 0xffffffffu);         // global_addr[95:64]
    g0.w = (unsigned)((ga >> 32) & 0x1ffffffu)   // global_addr[120:96]
         | (2u << 30);                           // type=2 ("image")

    v8i g1;
    g1[0] = (2 << 16);                 // wg_mask=0, data_size=2 (4 bytes)
    g1[1] = (64 << 16);                // tensor_dim0[15:0]=64 in bits[63:48]
    g1[2] = (64 << 16);                // dim0 hi=0 | tensor_dim1 lo16 = 64
    g1[3] = (64 << 16);                // dim1 hi=0 | tile_dim0 = 64
    g1[4] = (9 << 16) | 64;            // tile_dim1=64 | tile_dim2=9
    g1[5] = 576;                       // tensor_dim0_stride = 576 (lo32)
    g1[6] = (64 << 16);                // stride0 hi=0 | tensor_dim1_stride lo16=64
    g1[7] = 0;                         // stride1 hi32 = 0

    v4i g2;
    g2[0] = 9;                         // tensor_dim2 = 9
    g2[1] = 0;                         // tensor_dim3 = 0 (iterate off)
    g2[2] = 0;                         // tensor_dim2_stride = 0 (unused)
    g2[3] = 0;                         // tile_dim3 = 0
    v4i g3 = {0, 0, 0, 0};
    v8i g4 = {0, 0, 0, 0, 0, 0, 0, 0}; // extra operand of 6-arg form: zeros

    __builtin_amdgcn_tensor_load_to_lds(g0, g1, g2, g3, g4, /*cpol=*/0);
    __builtin_amdgcn_s_wait_tensorcnt(0);
  }
  if (tid < COUT) lb[tid] = bias[tid];
#else
  // ---- Fallback: manual transposed staging ----
  {
    const float4* __restrict__ src = (const float4*)w;
    float4* dst = (float4*)lw;
#pragma unroll
    for (int i = 0; i < (9 * CIN * COUT / 4) / 256; ++i) {
      const int f4 = tid + i * 256;
      const int k  = f4 >> 4;
      const int m4 = f4 & 15;
      const int c  = k / 9;
      const int t  = k - c * 9;
      dst[(t * CIN + c) * 16 + m4] = src[f4];
    }
    if (tid < COUT) lb[tid] = bias[tid];
  }
#endif
  __syncthreads();

  const int lane  = tid & 31;
  const int m     = lane & 15;       // A: M index / B,C: N index
  const int hi2   = (lane >> 4) * 2; // K-slot offset for this lane half: 0 or 2
  const int mbase = (wid & 3) * 16;  // this wave's cout tile base
  const int nwave = wid >> 2;        // 0..1: pixel sub-tile

  const int wg   = blockIdx.x;
  const int b    = wg / 49;                    // 49 WGs per batch image
  const int pix0 = (wg - b * 49) * PIX_PER_WG;

  const float* __restrict__ xb = x + (size_t)b * CIN * PIX;
  float* __restrict__ ob       = out + (size_t)b * COUT * PIX;

  const int abase_lane = hi2 * COUT + mbase + m;
  const float* __restrict__ xpix_c = xb + hi2 * PIX;  // channels hi2, hi2+1

  // One 16x16 accumulator tile per chunk; per-step WMMAs are independent.
  v8f acc[CHUNKS];
#pragma unroll
  for (int ch = 0; ch < CHUNKS; ++ch)
    acc[ch] = (v8f){0.f, 0.f, 0.f, 0.f, 0.f, 0.f, 0.f, 0.f};

  int ohs[CHUNKS], ows[CHUNKS];
#pragma unroll
  for (int ch = 0; ch < CHUNKS; ++ch) {
    const int pix = pix0 + ch * 32 + nwave * 16 + m;
    ohs[ch] = pix / HW;
    ows[ch] = pix - ohs[ch] * HW;
  }

#pragma unroll
  for (int t = 0; t < 9; ++t) {               // taps: dh,dw compile-time consts
    const int dh = (t / 3) - 1;
    const int dw = (t % 3) - 1;

    unsigned off[CHUNKS];
    bool     vld[CHUNKS];
#pragma unroll
    for (int ch = 0; ch < CHUNKS; ++ch) {
      const int ih = ohs[ch] + dh;
      const int iw = ows[ch] + dw;
      vld[ch] = ((unsigned)ih < HW) & ((unsigned)iw < HW);
      off[ch] = vld[ch] ? (unsigned)(ih * HW + iw) : 0u;
    }

    const int aoff_t = t * (CIN * COUT) + abase_lane;

#pragma unroll 4
    for (int s = 0; s < CIN / 4; ++s) {       // 16 K-steps of 4 channels
      // A operand: one ds_load_2addr worth, reused by all 8 chunk-WMMAs
      v2f av;
      av.x = lw[aoff_t + s * 4 * COUT];
      av.y = lw[aoff_t + s * 4 * COUT + COUT];

#pragma unroll
      for (int ch = 0; ch < CHUNKS; ++ch) {
        // B operand: channels 4s+hi2, 4s+hi2+1 at this lane's pixel
        const float r0 = xpix_c[off[ch] + (unsigned)(s * 4 * PIX)];
        const float r1 = xpix_c[off[ch] + (unsigned)(s * 4 * PIX + PIX)];
        v2f bmv;
        bmv.x = vld[ch] ? r0 : 0.f;
        bmv.y = vld[ch] ? r1 : 0.f;

        acc[ch] = __builtin_amdgcn_wmma_f32_16x16x4_f32(
            false, av, false, bmv, (short)0, acc[ch], false, false);
      }
    }
  }

  // ---- bias + store: C layout = VGPR r -> M = r + 8*(lane>=16), N = lane&15 ----
#pragma unroll
  for (int ch = 0; ch < CHUNKS; ++ch) {
    const int pix = pix0 + ch * 32 + nwave * 16 + m;
#pragma unroll
    for (int r = 0; r < 8; ++r) {
      const int mm = mbase + r + 4 * hi2;   // r + 8*hi
      ob[(size_t)mm * PIX + pix] = acc[ch][r] + lb[mm];
    }
  }
}

extern "C" void kernel_launch(void* const* d_in, const int* in_sizes, int n_in,
                              void* d_out, int out_size, void* d_ws, size_t ws_size,
                              hipStream_t stream) {
  const float* x    = (const float*)d_in[0];
  const float* w    = (const float*)d_in[1];
  const float* bias = (const float*)d_in[2];
  float* out        = (float*)d_out;

  dim3 grid(BATCH * (PIX / PIX_PER_WG));  // 16 * 49 = 784 workgroups
  dim3 block(256);
  conv3x3_im2col_wmma_f32<<<grid, block, 0, stream>>>(x, w, bias, out);
}